// TopicGenerator_23811298689835
// MI455X (gfx1250) — compile-verified
//
#include <hip/hip_runtime.h>
#include <hip/hip_bf16.h>
#include <math.h>

#define NEGV (-10000000000.0f)

typedef __attribute__((ext_vector_type(2))) float v2f;
typedef __attribute__((ext_vector_type(8))) float v8f;

// problem dims
constexpr int B  = 16;
constexpr int QL = 32;
constexpr int TN = 10;
constexpr int AL = 10;
constexpr int M  = TN * AL;     // 100 alias rows per batch
constexpr int E  = 300;
constexpr int EP = 304;         // padded embedding width (mult of 16)
constexpr int H  = 256;
constexpr int CW = 2 * E;       // enc_in width = 600
constexpr int MP = 112;         // per-batch alias rows padded to 7 tiles of 16
constexpr int ROWPAD = 1712;    // 16*107 >= 15*100 + 112 (tile overreach safe)

// workspace layout (floats)
constexpr size_t OFF_QEMB  = 0;                                  // B*QL*EP
constexpr size_t OFF_ENCIN = OFF_QEMB + (size_t)B * QL * EP;     // ROWPAD*CW  (cols 0..299 alig, 300..599 a_emb)
constexpr size_t OFF_ATT   = OFF_ENCIN + (size_t)ROWPAD * CW;    // B*MP*QL  masked pre-softmax atten
constexpr size_t OFF_ATT1  = OFF_ATT + (size_t)B * MP * QL;      // B*MP*QL  post-softmax
constexpr size_t OFF_AW1   = OFF_ATT1 + (size_t)B * MP * QL;     // B*MP     a_emb . W1
constexpr size_t OFF_QW2   = OFF_AW1 + (size_t)B * MP;           // B*QL     q_emb . W2
constexpr size_t OFF_MAXA  = OFF_QW2 + (size_t)B * QL;           // B*TN*QL  max over al
constexpr size_t OFF_SCORE = OFF_MAXA + (size_t)B * TN * QL;     // B*TN     compare @ qlin_w (pre-bias)

// ---------------------------------------------------------------------------
// K1: gather + L2-normalize embedding rows (one wave per row), fold in the
// rank-1 dot products (W1 for alias rows, W2 for query rows). Also zero the
// score accumulator.
// ---------------------------------------------------------------------------
__global__ void tg_gather_norm(const int* __restrict__ q, const int* __restrict__ alia,
                               const float* __restrict__ emb, const float* __restrict__ sim_w,
                               float* __restrict__ ws) {
  const int lane = threadIdx.x;
  const int row  = blockIdx.x;

  if (row == 0) {
    float* score = ws + OFF_SCORE;
    for (int i = lane; i < B * TN; i += 32) score[i] = 0.0f;
  }

  int idx, di;
  float* dst;
  const float* wvec;
  float* dots;
  bool is_q = (row < B * QL);
  if (is_q) {
    idx  = q[row];
    dst  = ws + OFF_QEMB + (size_t)row * EP;
    wvec = sim_w + E;            // W2
    dots = ws + OFF_QW2;
    di   = row;
  } else {
    int r = row - B * QL;        // 0..1599 flat alias row
    idx  = alia[r];
    dst  = ws + OFF_ENCIN + (size_t)r * CW + E;  // a_emb part of enc_in
    wvec = sim_w;                // W1
    dots = ws + OFF_AW1;
    di   = (r / M) * MP + (r % M);
  }

  const float* src = emb + (size_t)idx * E;
  float ss = 0.0f;
  for (int e = lane; e < E; e += 32) { float v = src[e]; ss += v * v; }
  for (int o = 16; o > 0; o >>= 1) ss += __shfl_xor(ss, o, 32);
  float inv = (idx == 0) ? 0.0f : 1.0f / fmaxf(sqrtf(ss), 1e-12f);

  float dot = 0.0f;
  for (int e = lane; e < E; e += 32) {
    float v = src[e] * inv;
    dst[e] = v;
    dot += v * wvec[e];
  }
  for (int o = 16; o > 0; o >>= 1) dot += __shfl_xor(dot, o, 32);
  if (lane == 0) dots[di] = dot;
  if (is_q && lane < EP - E) dst[E + lane] = 0.0f;   // pad cols 300..303
}

// ---------------------------------------------------------------------------
// K2: bilinear attention via WMMA f32 16x16x4.
//   atten[b,m,q] = sum_e a_emb[b,m,e]*W3[e]*q_emb[b,q,e] + aW1[m] + qW2[q] + sim_b
// then mask.  grid (ntiles=2, mtiles=7, B), one wave per 16x16 tile.
// ---------------------------------------------------------------------------
__global__ void tg_atten(const int* __restrict__ q, const int* __restrict__ alia,
                         const float* __restrict__ sim_w, const float* __restrict__ sim_b,
                         float* __restrict__ ws) {
  const int lane = threadIdx.x, half = lane >> 4, l16 = lane & 15;
  const int nt = blockIdx.x, mt = blockIdx.y, b = blockIdx.z;

  const float* A  = ws + OFF_ENCIN + ((size_t)b * M + mt * 16) * CW + E; // a_emb rows
  const float* qe = ws + OFF_QEMB + (size_t)b * QL * EP;
  const float* W3 = sim_w + 2 * E;

  v8f c = {};
  const int nq = nt * 16 + l16;                    // q column this lane owns in B-frag
  for (int k0 = 0; k0 < E; k0 += 4) {
    int ka = k0 + 2 * half;
    v2f a, bb;
    a.x  = A[(size_t)l16 * CW + ka]     * W3[ka];
    a.y  = A[(size_t)l16 * CW + ka + 1] * W3[ka + 1];
    bb.x = qe[(size_t)nq * EP + ka];
    bb.y = qe[(size_t)nq * EP + ka + 1];
    c = __builtin_amdgcn_wmma_f32_16x16x4_f32(false, a, false, bb, (short)0, c, false, false);
  }

  float qmask = (q[b * QL + nq] != 0) ? 1.0f : 0.0f;
  float qw2   = ws[OFF_QW2 + b * QL + nq];
  float sb    = sim_b[0];
  for (int r = 0; r < 8; ++r) {
    int m = mt * 16 + r + 8 * half;
    if (m < M) {
      float amask = (alia[b * M + m] != 0) ? 1.0f : 0.0f;
      float mask  = amask * qmask;
      float val   = c[r] + ws[OFF_AW1 + b * MP + m] + qw2 + sb;
      ws[OFF_ATT + ((size_t)b * MP + m) * QL + nq] = mask * val + (1.0f - mask) * NEGV;
    }
  }
}

// ---------------------------------------------------------------------------
// K3: softmax over ql=32 per (b,m) row (one wave per row).
// ---------------------------------------------------------------------------
__global__ void tg_softmax_rows(float* __restrict__ ws) {
  const int lane = threadIdx.x;
  const int row = blockIdx.x;            // 0..B*M-1
  const int b = row / M, m = row % M;
  const size_t base = ((size_t)b * MP + m) * QL;
  float x = ws[OFF_ATT + base + lane];
  float mx = x;
  for (int o = 16; o > 0; o >>= 1) mx = fmaxf(mx, __shfl_xor(mx, o, 32));
  float e = __expf(x - mx);
  float s = e;
  for (int o = 16; o > 0; o >>= 1) s += __shfl_xor(s, o, 32);
  ws[OFF_ATT1 + base + lane] = e / s;
}

// ---------------------------------------------------------------------------
// K3b: max over al of masked pre-softmax atten -> max_atten[b,t,q]
// ---------------------------------------------------------------------------
__global__ void tg_max_al(float* __restrict__ ws) {
  const int lane = threadIdx.x;
  const int blk = blockIdx.x;            // 0..B*TN-1
  const int b = blk / TN, t = blk % TN;
  float mx = ws[OFF_ATT + ((size_t)b * MP + t * AL) * QL + lane];
  for (int a = 1; a < AL; ++a)
    mx = fmaxf(mx, ws[OFF_ATT + ((size_t)b * MP + t * AL + a) * QL + lane]);
  ws[OFF_MAXA + (size_t)blk * QL + lane] = mx;
}

// ---------------------------------------------------------------------------
// K4: alig = atten1(100x32) @ q_emb(32x300) via WMMA f32 16x16x4.
// Writes into enc_in cols 0..299.  grid (ntiles=19, mtiles=7, B).
// ---------------------------------------------------------------------------
__global__ void tg_alig(float* __restrict__ ws) {
  const int lane = threadIdx.x, half = lane >> 4, l16 = lane & 15;
  const int nt = blockIdx.x, mt = blockIdx.y, b = blockIdx.z;

  const float* A  = ws + OFF_ATT1 + ((size_t)b * MP + mt * 16) * QL;
  const float* qe = ws + OFF_QEMB + (size_t)b * QL * EP;
  const int ne = nt * 16 + l16;                       // e column (< 304)

  v8f c = {};
  for (int k0 = 0; k0 < QL; k0 += 4) {
    int ka = k0 + 2 * half;
    v2f a, bb;
    a.x  = A[(size_t)l16 * QL + ka];
    a.y  = A[(size_t)l16 * QL + ka + 1];
    bb.x = qe[(size_t)ka * EP + ne];
    bb.y = qe[(size_t)(ka + 1) * EP + ne];
    c = __builtin_amdgcn_wmma_f32_16x16x4_f32(false, a, false, bb, (short)0, c, false, false);
  }
  if (ne < E) {
    for (int r = 0; r < 8; ++r) {
      int m = mt * 16 + r + 8 * half;
      if (m < M) ws[OFF_ENCIN + ((size_t)b * M + m) * CW + ne] = c[r];
    }
  }
}

// ---------------------------------------------------------------------------
// K5: enc = relu(enc_in(1600x600) @ enc_w(600x256) + enc_b), fused with the
// qlin_w reduction and the sum over al.  N is split across waves too (the
// qlin reduction is additive over disjoint n ranges), giving 16x100 = 1600
// independent waves instead of 100 -- this kernel dominates the FLOPs and the
// workload is latency-bound, so parallelism beats A-reuse (A is L2-resident).
// grid (ntiles=16, mtiles=100), one wave per 16x16 output tile, K = 150 WMMAs.
// ---------------------------------------------------------------------------
__global__ void tg_enc(const float* __restrict__ enc_w, const float* __restrict__ enc_b,
                       const float* __restrict__ qlin_w, float* __restrict__ ws) {
  const int lane = threadIdx.x, half = lane >> 4, l16 = lane & 15;
  const int nt = blockIdx.x;                       // 0..15
  const int mt = blockIdx.y;                       // 0..99
  const float* A = ws + OFF_ENCIN + (size_t)mt * 16 * CW;
  float* score = ws + OFF_SCORE;
  const int n = nt * 16 + l16;

  v8f c = {};
  for (int k0 = 0; k0 < CW; k0 += 4) {
    int ka = k0 + 2 * half;
    v2f a, bb;
    a.x  = A[(size_t)l16 * CW + ka];
    a.y  = A[(size_t)l16 * CW + ka + 1];
    bb.x = enc_w[(size_t)ka * H + n];
    bb.y = enc_w[(size_t)(ka + 1) * H + n];
    c = __builtin_amdgcn_wmma_f32_16x16x4_f32(false, a, false, bb, (short)0, c, false, false);
  }

  const float eb = enc_b[n], qw = qlin_w[n];
  for (int r = 0; r < 8; ++r) {
    float v = fmaxf(c[r] + eb, 0.0f) * qw;
    for (int o = 1; o <= 8; o <<= 1) v += __shfl_xor(v, o, 32);  // reduce 16 lanes in half
    if (l16 == 0) {
      int mg = mt * 16 + r + 8 * half;             // global row < 1600
      int b = mg / M, t = (mg % M) / AL;
      atomicAdd(&score[b * TN + t], v);
    }
  }
}

// ---------------------------------------------------------------------------
// K6: epilogue. logits, masked logits (output 1), softmax over TN, combine
// with max_atten, softmax over ql (output 2).  One wave per batch.
// ---------------------------------------------------------------------------
__global__ void tg_final(const float* __restrict__ els, const float* __restrict__ elo,
                         const float* __restrict__ cate, const float* __restrict__ qlin_b,
                         const float* __restrict__ cls_w, const float* __restrict__ cls_b,
                         const float* __restrict__ ws, float* __restrict__ out) {
  const int b = blockIdx.x, lane = threadIdx.x;
  __shared__ float sml[TN];

  if (lane < TN) {
    int i = b * TN + lane;
    float sc = ws[OFF_SCORE + i] + qlin_b[0];
    float lg = cls_b[0] + cls_w[0] * elo[i] + cls_w[1] * els[i] +
               cls_w[2] * cate[2 * i] + cls_w[3] * cate[2 * i + 1] + cls_w[4] * sc;
    float tm = (els[i] + elo[i] != 0.0f) ? 1.0f : 0.0f;
    float ml = tm * lg + (1.0f - tm) * NEGV;
    out[i] = ml;               // output 1: mask_logits (B, TN)
    sml[lane] = ml;
  }
  __syncthreads();

  float mx = sml[0];
  for (int t = 1; t < TN; ++t) mx = fmaxf(mx, sml[t]);
  float psum = 0.0f, p[TN];
  for (int t = 0; t < TN; ++t) { p[t] = __expf(sml[t] - mx); psum += p[t]; }
  float s = 0.0f;
  for (int t = 0; t < TN; ++t)
    s += (p[t] / psum) * ws[OFF_MAXA + ((size_t)b * TN + t) * QL + lane];
  float m2 = s;
  for (int o = 16; o > 0; o >>= 1) m2 = fmaxf(m2, __shfl_xor(m2, o, 32));
  float e = __expf(s - m2);
  float s2 = e;
  for (int o = 16; o > 0; o >>= 1) s2 += __shfl_xor(s2, o, 32);
  out[B * TN + b * QL + lane] = e / s2;   // output 2: atten_out (B, QL)
}

extern "C" void kernel_launch(void* const* d_in, const int* in_sizes, int n_in,
                              void* d_out, int out_size, void* d_ws, size_t ws_size,
                              hipStream_t stream) {
  const int*   q      = (const int*)d_in[0];
  const int*   alia   = (const int*)d_in[1];
  const float* els    = (const float*)d_in[2];
  const float* elo    = (const float*)d_in[3];
  const float* cate   = (const float*)d_in[4];
  const float* emb    = (const float*)d_in[5];
  const float* sim_w  = (const float*)d_in[6];
  const float* sim_b  = (const float*)d_in[7];
  const float* enc_w  = (const float*)d_in[8];
  const float* enc_b  = (const float*)d_in[9];
  const float* qlin_w = (const float*)d_in[10];
  const float* qlin_b = (const float*)d_in[11];
  const float* cls_w  = (const float*)d_in[12];
  const float* cls_b  = (const float*)d_in[13];
  float* out = (float*)d_out;
  float* ws  = (float*)d_ws;

  tg_gather_norm<<<B * QL + B * M, 32, 0, stream>>>(q, alia, emb, sim_w, ws);
  tg_atten<<<dim3(QL / 16, MP / 16, B), 32, 0, stream>>>(q, alia, sim_w, sim_b, ws);
  tg_softmax_rows<<<B * M, 32, 0, stream>>>(ws);
  tg_max_al<<<B * TN, 32, 0, stream>>>(ws);
  tg_alig<<<dim3(EP / 16, MP / 16, B), 32, 0, stream>>>(ws);
  tg_enc<<<dim3(H / 16, (B * M) / 16), 32, 0, stream>>>(enc_w, enc_b, qlin_w, ws);
  tg_final<<<B, 32, 0, stream>>>(els, elo, cate, qlin_b, cls_w, cls_b, ws, out);
}